// QuantizationObjective_2276332667063
// MI455X (gfx1250) — compile-verified
//
#include <hip/hip_runtime.h>
#include <hip/hip_bf16.h>

typedef __attribute__((ext_vector_type(2))) float v2f;
typedef __attribute__((ext_vector_type(8))) float v8f;

#define DIN 1024
#define ROWS_PER_BLK 16
#define KSTEPS 256   // 1024 / 4
#define WAVES 8
#define TPB 256

// ---------------------------------------------------------------------------
// Kernel 1: permute Sigma into WMMA-B-fragment order.
// B fragment layout for V_WMMA_F32_16X16X4_F32 (B is 4x16, 2 VGPRs):
//   VGPR0: lanes 0-15 -> K=k0,   lanes 16-31 -> K=k0+2  (col = lane&15)
//   VGPR1: lanes 0-15 -> K=k0+1, lanes 16-31 -> K=k0+3
// Stored so that one v2f (b64) load per lane fetches (VGPR0,VGPR1).
// ---------------------------------------------------------------------------
__global__ __launch_bounds__(256) void k_sigma_swizzle(const float* __restrict__ S,
                                                       v2f* __restrict__ out) {
    int tid  = blockIdx.x * blockDim.x + threadIdx.x;   // 0 .. 64*256*32-1
    int lane = tid & 31;
    int ks   = (tid >> 5) & (KSTEPS - 1);
    int nt   = tid >> 13;                               // N-tile 0..63
    int k0   = ks * 4 + ((lane >> 4) << 1);
    int col  = nt * 16 + (lane & 15);
    v2f v;
    v.x = S[k0 * DIN + col];
    v.y = S[(k0 + 1) * DIN + col];
    out[tid] = v;
}

// ---------------------------------------------------------------------------
// Kernel 2: fused dual-GEMM + reductions for one 16-row strip per workgroup.
// ---------------------------------------------------------------------------
__global__ __launch_bounds__(256) void k_main(const float* __restrict__ W,
                                              const float* __restrict__ Q,
                                              const float* __restrict__ s,
                                              const float* __restrict__ Theta,
                                              const v2f*  __restrict__ SigmaSw,
                                              float* __restrict__ q_part,
                                              float* __restrict__ p_part,
                                              float* __restrict__ b_part) {
    __shared__ float sDW[ROWS_PER_BLK * DIN];           // 64 KB: DeltaW strip

    const int blk  = blockIdx.x;
    const int tid  = threadIdx.x;
    const int lane = tid & 31;
    const int w    = tid >> 5;                          // wave id 0..7
    const long base = (long)blk * ROWS_PER_BLK * DIN;

    // ---- Stage DeltaW = W - s*(Q + Th) into LDS; accumulate binary reg ----
    float bsum = 0.0f;
    for (int e = tid; e < ROWS_PER_BLK * DIN; e += TPB) {
        int   row = e >> 10;
        float sv  = s[blk * ROWS_PER_BLK + row];
        float wv  = W[base + e];
        float qv  = Q[base + e];
        float tv  = tanhf(Theta[base + e]);
        float th  = (tv + 1.0f) * 0.6f - 0.1f;          // (tanh+1)/2*1.2 - 0.1
        th = fminf(fmaxf(th, 0.0f), 1.0f);
        sDW[e] = wv - sv * (qv + th);
        float u = 2.0f * th - 1.0f;
        bsum += 1.0f - u * u;                           // beta == 2
    }
    __syncthreads();

    for (int m = 16; m >= 1; m >>= 1) bsum += __shfl_xor(bsum, m);
    if (lane == 0) b_part[blk * WAVES + w] = bsum;

    // ---- A-fragment bases (A is 16x4: lanes 0-15 -> M, VGPR0/1 = K pairs) ----
    const int rowA = lane & 15;
    const int koff = (lane >> 4) << 1;                  // 0 or 2
    const float* Wfrag  = W + base + rowA * DIN + koff;
    const float* DWfrag = sDW + rowA * DIN + koff;

    float qa[8];
#pragma unroll
    for (int v = 0; v < 8; ++v) qa[v] = 0.0f;
    float p = 0.0f;

    // ---- Each wave owns 8 N-tiles; dual WMMA accumulation over full K ----
    for (int t = 0; t < 8; ++t) {
        int n = w * 8 + t;
        v8f accW = {};
        v8f accD = {};
        const v2f* Bp = SigmaSw + (size_t)(n * KSTEPS) * 32 + lane;
#pragma unroll 4
        for (int ks = 0; ks < KSTEPS; ++ks) {
            v2f bfrag = Bp[ks * 32];
            v2f aW = *(const v2f*)(Wfrag  + ks * 4);
            v2f aD = *(const v2f*)(DWfrag + ks * 4);
            accW = __builtin_amdgcn_wmma_f32_16x16x4_f32(
                false, aW, false, bfrag, (short)0, accW, false, false);
            accD = __builtin_amdgcn_wmma_f32_16x16x4_f32(
                false, aD, false, bfrag, (short)0, accD, false, false);
        }
        // D layout: VGPR v, lanes 0-15 -> row v, lanes 16-31 -> row v+8; col = lane&15
        int colD  = n * 16 + (lane & 15);
        int rbase = (lane >> 4) << 3;
#pragma unroll
        for (int v = 0; v < 8; ++v) {
            float dw = sDW[(rbase + v) * DIN + colD];
            p     += accD[v] * dw;                      // activation partial
            qa[v] += accW[v] * dw;                      // geom per-row partial
        }
    }

    // ---- per-row geom partials: reduce across 16-lane halves ----
#pragma unroll
    for (int v = 0; v < 8; ++v) {
        float qv = qa[v];
        for (int m = 1; m < 16; m <<= 1) qv += __shfl_xor(qv, m);
        if ((lane & 15) == 0) {
            int row = blk * ROWS_PER_BLK + ((lane >> 4) << 3) + v;
            q_part[row * WAVES + w] = qv;
        }
    }
    // ---- activation partial: full-wave reduce ----
    for (int m = 16; m >= 1; m >>= 1) p += __shfl_xor(p, m);
    if (lane == 0) p_part[blk * WAVES + w] = p;
}

// ---------------------------------------------------------------------------
// Kernel 3: deterministic final fold.
// ---------------------------------------------------------------------------
__global__ __launch_bounds__(256) void k_final(const float* __restrict__ q_part,
                                               const float* __restrict__ p_part,
                                               const float* __restrict__ b_part,
                                               float* __restrict__ out,
                                               int nrows, int nwg) {
    __shared__ float rp[WAVES], rb[WAVES], rg[WAVES];
    int tid = threadIdx.x, lane = tid & 31, w = tid >> 5;
    float ap = 0.0f, ab = 0.0f, ag = 0.0f;
    int npb = nwg * WAVES;
    for (int i = tid; i < npb; i += TPB) { ap += p_part[i]; ab += b_part[i]; }
    for (int i = tid; i < nrows; i += TPB) {
        float q = 0.0f;
#pragma unroll
        for (int j = 0; j < WAVES; ++j) q += q_part[i * WAVES + j];
        ag += q * q;
    }
    for (int m = 16; m >= 1; m >>= 1) {
        ap += __shfl_xor(ap, m);
        ab += __shfl_xor(ab, m);
        ag += __shfl_xor(ag, m);
    }
    if (lane == 0) { rp[w] = ap; rb[w] = ab; rg[w] = ag; }
    __syncthreads();
    if (tid == 0) {
        float P = 0.0f, B = 0.0f, G = 0.0f;
        for (int j = 0; j < WAVES; ++j) { P += rp[j]; B += rb[j]; G += rg[j]; }
        // total = activation + lamb_reg*binary + lamb * sum((2q)^2)
        out[0] = P + 0.0002f * B + 0.05f * 4.0f * G;
    }
}

extern "C" void kernel_launch(void* const* d_in, const int* in_sizes, int n_in,
                              void* d_out, int out_size, void* d_ws, size_t ws_size,
                              hipStream_t stream) {
    const float* W     = (const float*)d_in[0];
    const float* Sigma = (const float*)d_in[1];
    const float* Q     = (const float*)d_in[2];
    const float* s     = (const float*)d_in[3];
    const float* Theta = (const float*)d_in[4];

    const int d_out_rows = 32768;
    const int nblk = d_out_rows / ROWS_PER_BLK;         // 2048

    char* ws = (char*)d_ws;
    v2f*   SigmaSw = (v2f*)ws;                                      // 4 MB
    float* q_part  = (float*)(ws + (size_t)4 * 1024 * 1024);        // 1 MB
    float* p_part  = (float*)(ws + (size_t)5 * 1024 * 1024);        // 64 KB
    float* b_part  = (float*)(ws + (size_t)5 * 1024 * 1024 + 65536);// 64 KB

    hipLaunchKernelGGL(k_sigma_swizzle, dim3(2048), dim3(256), 0, stream,
                       Sigma, SigmaSw);
    hipLaunchKernelGGL(k_main, dim3(nblk), dim3(256), 0, stream,
                       W, Q, s, Theta, SigmaSw, q_part, p_part, b_part);
    hipLaunchKernelGGL(k_final, dim3(1), dim3(256), 0, stream,
                       q_part, p_part, b_part, (float*)d_out, d_out_rows, nblk);
}